// MACE_24404004176562
// MI455X (gfx1250) — compile-verified
//
#include <hip/hip_runtime.h>
#include <math.h>

// ---------------------------------------------------------------------------
// MACE energy + forces for MI455X (gfx1250), f32 WMMA (16x16x4) throughout.
//
// Input pointer mapping (setup_inputs() dict order; params pytree flattened
// with sorted dict keys, list in order):
//   0 positions[N,3] 1 node_attrs[N,Z] 2 shifts[E,3]
//   3 W_embed[Z,32]  4 atomic_energies[Z]
//   5..17  layer0: W_msg[3,32,32] W_prod[32,32] W_sc[Z,32,32] W_up[32,32]
//                  m1[8,64] m2[64,64] m3[64,96] p1[Z,32] p2[Z,32,3] p3[Z,32,3]
//                  r0[32,1] rA[32,16] rB[16,1]
//   18..30 layer1: same
//   31 senders[E] 32 receivers[E] 33 batch[N]
// Output: total[8] ++ node_E[N] ++ forces[N,3]   (float32)
// ---------------------------------------------------------------------------

#define KC 32   // channels
#define LMX 9   // spherical-harmonic components (l<=2)

typedef __attribute__((ext_vector_type(2))) float v2f;
typedef __attribute__((ext_vector_type(8))) float v8f;

__device__ __forceinline__ v8f wmma4(v2f a, v2f b, v8f c) {
  // D(16x16) = A(16x4) * B(4x16) + C   -- full f32 precision matrix op
  return __builtin_amdgcn_wmma_f32_16x16x4_f32(false, a, false, b, (short)0, c,
                                               false, false);
}
__device__ __forceinline__ v8f zero8() {
  v8f z = {0.f, 0.f, 0.f, 0.f, 0.f, 0.f, 0.f, 0.f};
  return z;
}
__device__ __forceinline__ float siluf(float x) { return x / (1.f + __expf(-x)); }
__device__ __forceinline__ float dsiluf(float x) {
  float s = 1.f / (1.f + __expf(-x));
  return s * (1.f + x * (1.f - s));
}

// ---------------------------------------------------------------------------
// Edge geometry forward: vec, r, Y (9 real SH, component norm), ef (8 Bessel*cut)
// ---------------------------------------------------------------------------
__global__ void k_geom_fwd(const float* __restrict__ pos,
                           const float* __restrict__ shifts,
                           const int* __restrict__ snd, const int* __restrict__ rcv,
                           float* __restrict__ vec, float* __restrict__ rr,
                           float* __restrict__ Ysh, float* __restrict__ efb, int E) {
  long e = (long)blockIdx.x * blockDim.x + threadIdx.x;
  if (e >= E) return;
  int s = snd[e], d = rcv[e];
  float vx = pos[(long)d * 3 + 0] - pos[(long)s * 3 + 0] + shifts[e * 3 + 0];
  float vy = pos[(long)d * 3 + 1] - pos[(long)s * 3 + 1] + shifts[e * 3 + 1];
  float vz = pos[(long)d * 3 + 2] - pos[(long)s * 3 + 2] + shifts[e * 3 + 2];
  float r2 = vx * vx + vy * vy + vz * vz;
  float r = sqrtf(r2 + 1e-12f);
  vec[e * 3 + 0] = vx; vec[e * 3 + 1] = vy; vec[e * 3 + 2] = vz;
  rr[e] = r;
  float ir = 1.f / r;
  float x = vx * ir, y = vy * ir, z = vz * ir;
  const float s3 = 1.7320508075688772f, s5 = 2.2360679774997896f,
              s15 = 3.872983346207417f;
  Ysh[e * 9 + 0] = 1.f;
  Ysh[e * 9 + 1] = s3 * x;
  Ysh[e * 9 + 2] = s3 * y;
  Ysh[e * 9 + 3] = s3 * z;
  Ysh[e * 9 + 4] = s15 * x * y;
  Ysh[e * 9 + 5] = s15 * y * z;
  Ysh[e * 9 + 6] = 0.5f * s5 * (3.f * z * z - 1.f);
  Ysh[e * 9 + 7] = s15 * x * z;
  Ysh[e * 9 + 8] = 0.5f * s15 * (x * x - y * y);
  float rm = fmaxf(r, 1e-6f);
  float xc = r * 0.2f;  // r / R_MAX
  float cut = 0.f;
  if (xc < 1.f) {
    float x2 = xc * xc, x4 = x2 * x2, x5 = x4 * xc, x6 = x5 * xc, x7 = x6 * xc;
    cut = 1.f - 21.f * x5 + 35.f * x6 - 15.f * x7;  // p=5 polynomial cutoff
  }
  const float pref = 0.6324555320336759f;  // sqrt(2/5)
  const float PIf = 3.14159265358979323846f;
#pragma unroll
  for (int nn = 1; nn <= 8; ++nn)
    efb[e * 8 + nn - 1] = pref * sinf((float)nn * PIf * r * 0.2f) / rm * cut;
}

// ---------------------------------------------------------------------------
// Node embedding: h0 = W_embed[elem], node_E = atomic_energies[elem]
// ---------------------------------------------------------------------------
__global__ void k_embed(const float* __restrict__ attrs,
                        const float* __restrict__ Wemb,
                        const float* __restrict__ AE, int* __restrict__ elem,
                        float* __restrict__ h0, float* __restrict__ nodeE, int N,
                        int Zn) {
  long n = (long)blockIdx.x * blockDim.x + threadIdx.x;
  if (n >= N) return;
  int z = 0;
  float best = -1.f;
  for (int q = 0; q < Zn; ++q) {
    float a = attrs[n * Zn + q];
    if (a > best) { best = a; z = q; }
  }
  elem[n] = z;
  for (int k = 0; k < KC; ++k) h0[n * KC + k] = Wemb[z * KC + k];
  nodeE[n] = AE[z];
}

// ---------------------------------------------------------------------------
// Generic 32-wide node GEMM:  D[N,32] (=|+=) S[N,32] @ W  (W 32x32, opt. W^T)
// One WMMA wave per 16-node tile; 8 k-chunks x 2 n-tiles = 16 WMMAs.
// ---------------------------------------------------------------------------
__global__ __launch_bounds__(256) void k_gemm32(const float* __restrict__ S,
                                                const float* __restrict__ W,
                                                float* __restrict__ D, int N,
                                                int transB, int accum) {
  const int wave = threadIdx.x >> 5, lane = threadIdx.x & 31;
  const int row = lane & 15, kh = lane >> 4;
  const long n0 = ((long)blockIdx.x * (blockDim.x >> 5) + wave) * 16;
  if (n0 >= N) return;
  long nr = n0 + row;
  if (nr >= N) nr = N - 1;
#pragma unroll
  for (int nt = 0; nt < 2; ++nt) {
    v8f c;
#pragma unroll
    for (int v = 0; v < 8; ++v) {
      long mr = n0 + v + 8 * kh;
      c[v] = (accum && mr < N) ? D[mr * KC + nt * 16 + row] : 0.f;
    }
#pragma unroll
    for (int kc = 0; kc < 8; ++kc) {
      const int k0 = kc * 4 + kh * 2;
      v2f a; a.x = S[nr * KC + k0]; a.y = S[nr * KC + k0 + 1];
      v2f b;
      if (transB) {
        b.x = W[(nt * 16 + row) * KC + k0];
        b.y = W[(nt * 16 + row) * KC + k0 + 1];
      } else {
        b.x = W[k0 * KC + nt * 16 + row];
        b.y = W[(k0 + 1) * KC + nt * 16 + row];
      }
      c = wmma4(a, b, c);
    }
#pragma unroll
    for (int v = 0; v < 8; ++v) {
      long mr = n0 + v + 8 * kh;
      if (mr < N) D[mr * KC + nt * 16 + row] = c[v];
    }
  }
}

// Same, looping the 9 SH planes with per-plane W_msg[l_of_lm[m]].
__global__ __launch_bounds__(256) void k_gemm32_planes(
    const float* __restrict__ S, const float* __restrict__ Wmsg,
    float* __restrict__ D, int N, int transB) {
  const int wave = threadIdx.x >> 5, lane = threadIdx.x & 31;
  const int row = lane & 15, kh = lane >> 4;
  const long n0 = ((long)blockIdx.x * (blockDim.x >> 5) + wave) * 16;
  if (n0 >= N) return;
  long nr = n0 + row;
  if (nr >= N) nr = N - 1;
  const int LofLM[9] = {0, 1, 1, 1, 2, 2, 2, 2, 2};
  for (int m = 0; m < 9; ++m) {
    const float* Sp = S + (long)m * N * KC;
    const float* W = Wmsg + LofLM[m] * KC * KC;
    float* Dp = D + (long)m * N * KC;
#pragma unroll
    for (int nt = 0; nt < 2; ++nt) {
      v8f c = zero8();
#pragma unroll
      for (int kc = 0; kc < 8; ++kc) {
        const int k0 = kc * 4 + kh * 2;
        v2f a; a.x = Sp[nr * KC + k0]; a.y = Sp[nr * KC + k0 + 1];
        v2f b;
        if (transB) {
          b.x = W[(nt * 16 + row) * KC + k0];
          b.y = W[(nt * 16 + row) * KC + k0 + 1];
        } else {
          b.x = W[k0 * KC + nt * 16 + row];
          b.y = W[(k0 + 1) * KC + nt * 16 + row];
        }
        c = wmma4(a, b, c);
      }
#pragma unroll
      for (int v = 0; v < 8; ++v) {
        long mr = n0 + v + 8 * kh;
        if (mr < N) Dp[mr * KC + nt * 16 + row] = c[v];
      }
    }
  }
}

// ---------------------------------------------------------------------------
// Edge forward: radial MLP (WMMA f32) fused with message formation + atomic
// segment-sum into agg planes [m][N][32].  One wave = one 16-edge tile.
// ---------------------------------------------------------------------------
__global__ __launch_bounds__(64) void k_edge_fwd(
    const float* __restrict__ efb, const float* __restrict__ Ysh,
    const float* __restrict__ hup, const int* __restrict__ snd,
    const int* __restrict__ rcv, const float* __restrict__ m1,
    const float* __restrict__ m2, const float* __restrict__ m3,
    float* __restrict__ agg, int E, int N) {
  __shared__ float Tb[2][1024];    // t1 then t2 (16x64)
  __shared__ float RWb[2][1536];   // rw (16x96)
  const int wave = threadIdx.x >> 5, lane = threadIdx.x & 31;
  const int row = lane & 15, kh = lane >> 4;
  const long e0 = ((long)blockIdx.x * 2 + wave) * 16;
  if (e0 >= E) return;
  long er = e0 + row;
  if (er >= E) er = E - 1;
  float* T = Tb[wave];
  float* RW = RWb[wave];

  // GEMM1: t1 = silu(ef(16x8) @ m1(8x64))
#pragma unroll
  for (int nt = 0; nt < 4; ++nt) {
    v8f c = zero8();
#pragma unroll
    for (int kc = 0; kc < 2; ++kc) {
      const int k0 = kc * 4 + kh * 2;
      v2f a; a.x = efb[er * 8 + k0]; a.y = efb[er * 8 + k0 + 1];
      v2f b; b.x = m1[k0 * 64 + nt * 16 + row]; b.y = m1[(k0 + 1) * 64 + nt * 16 + row];
      c = wmma4(a, b, c);
    }
#pragma unroll
    for (int v = 0; v < 8; ++v) T[(v + 8 * kh) * 64 + nt * 16 + row] = siluf(c[v]);
  }
  // GEMM2: t2 = silu(t1 @ m2(64x64))  (in-place: all reads before stores)
  {
    v8f c[4];
#pragma unroll
    for (int nt = 0; nt < 4; ++nt) c[nt] = zero8();
#pragma unroll
    for (int kc = 0; kc < 16; ++kc) {
      const int k0 = kc * 4 + kh * 2;
      v2f a; a.x = T[row * 64 + k0]; a.y = T[row * 64 + k0 + 1];
#pragma unroll
      for (int nt = 0; nt < 4; ++nt) {
        v2f b; b.x = m2[k0 * 64 + nt * 16 + row]; b.y = m2[(k0 + 1) * 64 + nt * 16 + row];
        c[nt] = wmma4(a, b, c[nt]);
      }
    }
#pragma unroll
    for (int nt = 0; nt < 4; ++nt)
#pragma unroll
      for (int v = 0; v < 8; ++v)
        T[(v + 8 * kh) * 64 + nt * 16 + row] = siluf(c[nt][v]);
  }
  // GEMM3: rw = t2 @ m3(64x96)
  {
    v8f c[6];
#pragma unroll
    for (int nt = 0; nt < 6; ++nt) c[nt] = zero8();
#pragma unroll
    for (int kc = 0; kc < 16; ++kc) {
      const int k0 = kc * 4 + kh * 2;
      v2f a; a.x = T[row * 64 + k0]; a.y = T[row * 64 + k0 + 1];
#pragma unroll
      for (int nt = 0; nt < 6; ++nt) {
        v2f b; b.x = m3[k0 * 96 + nt * 16 + row]; b.y = m3[(k0 + 1) * 96 + nt * 16 + row];
        c[nt] = wmma4(a, b, c[nt]);
      }
    }
#pragma unroll
    for (int nt = 0; nt < 6; ++nt)
#pragma unroll
      for (int v = 0; v < 8; ++v) RW[(v + 8 * kh) * 96 + nt * 16 + row] = c[nt][v];
  }
  // Messages: m[e,k,m9] = rw[e,k,l(m9)] * hup[send,k] * Y[e,m9]; atomic segsum/16.
  const float inv = 1.f / 16.f;
  for (int e = 0; e < 16; ++e) {
    const long ge = e0 + e;
    if (ge >= E) break;
    const int s = snd[ge], d = rcv[ge];
    const float hv = hup[(long)s * KC + lane];
    const float w0 = RW[e * 96 + lane * 3 + 0] * hv * inv;
    const float w1 = RW[e * 96 + lane * 3 + 1] * hv * inv;
    const float w2 = RW[e * 96 + lane * 3 + 2] * hv * inv;
#pragma unroll
    for (int m = 0; m < 9; ++m) {
      const float wl = (m == 0) ? w0 : ((m < 4) ? w1 : w2);
      atomicAdd(&agg[((long)m * N + d) * KC + lane], wl * Ysh[ge * 9 + m]);
    }
  }
}

// ---------------------------------------------------------------------------
// Node pointwise forward: product basis B, sc, h' = B@W_prod + sc, readout.
// One wave per node, lane = channel.
// ---------------------------------------------------------------------------
template <int LAYER>
__global__ __launch_bounds__(256) void k_node_fwd(
    const float* __restrict__ h_in, const float* __restrict__ Ap,
    const int* __restrict__ elem, const float* __restrict__ Wsc,
    const float* __restrict__ Wprod, const float* __restrict__ p1,
    const float* __restrict__ p2, const float* __restrict__ p3,
    const float* __restrict__ r0, const float* __restrict__ rA,
    const float* __restrict__ rB, float* __restrict__ h_out,
    float* __restrict__ nodeE, int N) {
  const int wave = threadIdx.x >> 5, lane = threadIdx.x & 31;
  const long n = (long)blockIdx.x * (blockDim.x >> 5) + wave;
  if (n >= N) return;
  const int z = elem[n];
  float Am[9];
#pragma unroll
  for (int m = 0; m < 9; ++m) Am[m] = Ap[((long)m * N + n) * KC + lane];
  const float q0 = Am[0] * Am[0];
  const float q1 = Am[1] * Am[1] + Am[2] * Am[2] + Am[3] * Am[3];
  const float q2 = Am[4] * Am[4] + Am[5] * Am[5] + Am[6] * Am[6] +
                   Am[7] * Am[7] + Am[8] * Am[8];
  const float* p2z = p2 + ((long)z * KC + lane) * 3;
  const float* p3z = p3 + ((long)z * KC + lane) * 3;
  const float S2 = p2z[0] * q0 + p2z[1] * q1 + p2z[2] * q2;
  const float S3 = p3z[0] * q0 + p3z[1] * q1 + p3z[2] * q2;
  const float B = p1[(long)z * KC + lane] * Am[0] + S2 + Am[0] * S3;
  const float hv = h_in[n * KC + lane];
  float acc = 0.f;
  for (int k = 0; k < KC; ++k) {
    const float Bk = __shfl(B, k, 32);
    const float hk = __shfl(hv, k, 32);
    acc += Bk * Wprod[k * KC + lane] + hk * Wsc[((long)z * KC + k) * KC + lane];
  }
  h_out[n * KC + lane] = acc;
  float e;
  if (LAYER == 0) {
    float p = acc * r0[lane];
    for (int off = 16; off; off >>= 1) p += __shfl_xor(p, off, 32);
    e = p;
  } else {
    float zt = 0.f;
    for (int j = 0; j < KC; ++j) {
      const float hj = __shfl(acc, j, 32);
      if (lane < 16) zt += hj * rA[j * 16 + lane];
    }
    float pe = 0.f;
    if (lane < 16) pe = siluf(zt) * rB[lane];
    for (int off = 16; off; off >>= 1) pe += __shfl_xor(pe, off, 32);
    e = pe;
  }
  if (lane == 0) nodeE[n] += e;
}

__global__ void k_total(const float* __restrict__ nodeE,
                        const int* __restrict__ batch,
                        float* __restrict__ total, int N) {
  long n = (long)blockIdx.x * blockDim.x + threadIdx.x;
  if (n < N) atomicAdd(&total[batch[n]], nodeE[n]);
}

// ---------------------------------------------------------------------------
// Backward: readout1 gradient dS/dh2
// ---------------------------------------------------------------------------
__global__ __launch_bounds__(256) void k_readout1_bwd(
    const float* __restrict__ h2, const float* __restrict__ rA,
    const float* __restrict__ rB, float* __restrict__ gout, int N) {
  const int wave = threadIdx.x >> 5, lane = threadIdx.x & 31;
  const long n = (long)blockIdx.x * (blockDim.x >> 5) + wave;
  if (n >= N) return;
  const float hv = h2[n * KC + lane];
  float zt = 0.f;
  for (int j = 0; j < KC; ++j) {
    const float hj = __shfl(hv, j, 32);
    if (lane < 16) zt += hj * rA[j * 16 + lane];
  }
  float d = 0.f;
  if (lane < 16) d = dsiluf(zt) * rB[lane];
  float g = 0.f;
  for (int t = 0; t < 16; ++t) {
    const float dt = __shfl(d, t, 32);
    g += rA[lane * 16 + t] * dt;
  }
  gout[n * KC + lane] = g;
}

// ---------------------------------------------------------------------------
// Node pointwise backward: gout -> gA planes, g_h (sc path).
// ---------------------------------------------------------------------------
__global__ __launch_bounds__(256) void k_node_bwd(
    const float* __restrict__ gout_in, const float* __restrict__ addvec,
    const float* __restrict__ Ap, const int* __restrict__ elem,
    const float* __restrict__ Wsc, const float* __restrict__ Wprod,
    const float* __restrict__ p1, const float* __restrict__ p2,
    const float* __restrict__ p3, float* __restrict__ gA,
    float* __restrict__ g_h, int N, int write_gh) {
  const int wave = threadIdx.x >> 5, lane = threadIdx.x & 31;
  const long n = (long)blockIdx.x * (blockDim.x >> 5) + wave;
  if (n >= N) return;
  const int z = elem[n];
  const float go = gout_in[n * KC + lane] + (addvec ? addvec[lane] : 0.f);
  float gB = 0.f, gh = 0.f;
  for (int j = 0; j < KC; ++j) {
    const float gj = __shfl(go, j, 32);
    gB += gj * Wprod[lane * KC + j];
    gh += gj * Wsc[((long)z * KC + lane) * KC + j];
  }
  float Am[9];
#pragma unroll
  for (int m = 0; m < 9; ++m) Am[m] = Ap[((long)m * N + n) * KC + lane];
  const float q0 = Am[0] * Am[0];
  const float q1 = Am[1] * Am[1] + Am[2] * Am[2] + Am[3] * Am[3];
  const float q2 = Am[4] * Am[4] + Am[5] * Am[5] + Am[6] * Am[6] +
                   Am[7] * Am[7] + Am[8] * Am[8];
  const float* p2z = p2 + ((long)z * KC + lane) * 3;
  const float* p3z = p3 + ((long)z * KC + lane) * 3;
  const float S3 = p3z[0] * q0 + p3z[1] * q1 + p3z[2] * q2;
  const float gA0 = gB * (p1[(long)z * KC + lane] + S3);
  const float gn0 = gB * (p2z[0] + Am[0] * p3z[0]);
  const float gn1 = gB * (p2z[1] + Am[0] * p3z[1]);
  const float gn2 = gB * (p2z[2] + Am[0] * p3z[2]);
  gA[((long)0 * N + n) * KC + lane] = 2.f * Am[0] * gn0 + gA0;
#pragma unroll
  for (int m = 1; m < 4; ++m)
    gA[((long)m * N + n) * KC + lane] = 2.f * Am[m] * gn1;
#pragma unroll
  for (int m = 4; m < 9; ++m)
    gA[((long)m * N + n) * KC + lane] = 2.f * Am[m] * gn2;
  if (write_gh) g_h[n * KC + lane] = gh;
}

// ---------------------------------------------------------------------------
// Edge backward: recompute MLP (WMMA), message backward (ghup atomics, gY),
// then MLP backprop with transposed-B WMMA GEMMs -> gef.
// ---------------------------------------------------------------------------
__global__ __launch_bounds__(64) void k_edge_bwd(
    const float* __restrict__ efb, const float* __restrict__ Ysh,
    const float* __restrict__ hup, const int* __restrict__ snd,
    const int* __restrict__ rcv, const float* __restrict__ m1,
    const float* __restrict__ m2, const float* __restrict__ m3,
    const float* __restrict__ gagg, float* __restrict__ ghup,
    float* __restrict__ gY, float* __restrict__ gef, int E, int N,
    int need_ghup) {
  __shared__ float D1b[2][1024];  // silu'(t0)
  __shared__ float T1b[2][1024];  // silu(t0)
  __shared__ float D2b[2][1024];  // silu'(t2pre), later g_t0
  __shared__ float TBb[2][1024];  // t2, later g_t2v
  __shared__ float RWb[2][1536];  // rw, later grw (in-place)
  const int wave = threadIdx.x >> 5, lane = threadIdx.x & 31;
  const int row = lane & 15, kh = lane >> 4;
  const long e0 = ((long)blockIdx.x * 2 + wave) * 16;
  if (e0 >= E) return;
  long er = e0 + row;
  if (er >= E) er = E - 1;
  float *D1 = D1b[wave], *T1 = T1b[wave], *D2 = D2b[wave], *TB = TBb[wave],
        *RW = RWb[wave];

  // ---- forward recompute -------------------------------------------------
#pragma unroll
  for (int nt = 0; nt < 4; ++nt) {
    v8f c = zero8();
#pragma unroll
    for (int kc = 0; kc < 2; ++kc) {
      const int k0 = kc * 4 + kh * 2;
      v2f a; a.x = efb[er * 8 + k0]; a.y = efb[er * 8 + k0 + 1];
      v2f b; b.x = m1[k0 * 64 + nt * 16 + row]; b.y = m1[(k0 + 1) * 64 + nt * 16 + row];
      c = wmma4(a, b, c);
    }
#pragma unroll
    for (int v = 0; v < 8; ++v) {
      const int idx = (v + 8 * kh) * 64 + nt * 16 + row;
      T1[idx] = siluf(c[v]);
      D1[idx] = dsiluf(c[v]);
    }
  }
  {
    v8f c[4];
#pragma unroll
    for (int nt = 0; nt < 4; ++nt) c[nt] = zero8();
#pragma unroll
    for (int kc = 0; kc < 16; ++kc) {
      const int k0 = kc * 4 + kh * 2;
      v2f a; a.x = T1[row * 64 + k0]; a.y = T1[row * 64 + k0 + 1];
#pragma unroll
      for (int nt = 0; nt < 4; ++nt) {
        v2f b; b.x = m2[k0 * 64 + nt * 16 + row]; b.y = m2[(k0 + 1) * 64 + nt * 16 + row];
        c[nt] = wmma4(a, b, c[nt]);
      }
    }
#pragma unroll
    for (int nt = 0; nt < 4; ++nt)
#pragma unroll
      for (int v = 0; v < 8; ++v) {
        const int idx = (v + 8 * kh) * 64 + nt * 16 + row;
        TB[idx] = siluf(c[nt][v]);
        D2[idx] = dsiluf(c[nt][v]);
      }
  }
  {
    v8f c[6];
#pragma unroll
    for (int nt = 0; nt < 6; ++nt) c[nt] = zero8();
#pragma unroll
    for (int kc = 0; kc < 16; ++kc) {
      const int k0 = kc * 4 + kh * 2;
      v2f a; a.x = TB[row * 64 + k0]; a.y = TB[row * 64 + k0 + 1];
#pragma unroll
      for (int nt = 0; nt < 6; ++nt) {
        v2f b; b.x = m3[k0 * 96 + nt * 16 + row]; b.y = m3[(k0 + 1) * 96 + nt * 16 + row];
        c[nt] = wmma4(a, b, c[nt]);
      }
    }
#pragma unroll
    for (int nt = 0; nt < 6; ++nt)
#pragma unroll
      for (int v = 0; v < 8; ++v) RW[(v + 8 * kh) * 96 + nt * 16 + row] = c[nt][v];
  }

  // ---- message backward --------------------------------------------------
  const float inv = 1.f / 16.f;
  for (int e = 0; e < 16; ++e) {
    const long ge = e0 + e;
    if (ge >= E) break;
    const int s = snd[ge], d = rcv[ge];
    const float hv = hup[(long)s * KC + lane];
    const float w0 = RW[e * 96 + lane * 3 + 0];
    const float w1 = RW[e * 96 + lane * 3 + 1];
    const float w2 = RW[e * 96 + lane * 3 + 2];
    float gw0 = 0.f, gw1 = 0.f, gw2 = 0.f, ghv = 0.f;
    float gyp[9];
#pragma unroll
    for (int m = 0; m < 9; ++m) {
      const float gm = gagg[((long)m * N + d) * KC + lane] * inv;
      const float yv = Ysh[ge * 9 + m];
      const float wl = (m == 0) ? w0 : ((m < 4) ? w1 : w2);
      const float gmy = gm * yv;
      if (m == 0) gw0 += gmy * hv;
      else if (m < 4) gw1 += gmy * hv;
      else gw2 += gmy * hv;
      ghv += gmy * wl;
      gyp[m] = gm * wl * hv;
    }
    RW[e * 96 + lane * 3 + 0] = gw0;  // grw in-place (per (e,lane) cols)
    RW[e * 96 + lane * 3 + 1] = gw1;
    RW[e * 96 + lane * 3 + 2] = gw2;
    if (need_ghup) atomicAdd(&ghup[(long)s * KC + lane], ghv);
#pragma unroll
    for (int m = 0; m < 9; ++m) {
      float v = gyp[m];
      for (int off = 16; off; off >>= 1) v += __shfl_xor(v, off, 32);
      if (lane == 0) gY[ge * 9 + m] += v;
    }
  }

  // ---- MLP backward ------------------------------------------------------
  // g_t2v = (grw @ m3^T) * silu'(t2pre)  -> TB
  {
    v8f c[4];
#pragma unroll
    for (int nt = 0; nt < 4; ++nt) c[nt] = zero8();
#pragma unroll
    for (int kc = 0; kc < 24; ++kc) {
      const int k0 = kc * 4 + kh * 2;
      v2f a; a.x = RW[row * 96 + k0]; a.y = RW[row * 96 + k0 + 1];
#pragma unroll
      for (int nt = 0; nt < 4; ++nt) {
        v2f b; b.x = m3[(nt * 16 + row) * 96 + k0]; b.y = m3[(nt * 16 + row) * 96 + k0 + 1];
        c[nt] = wmma4(a, b, c[nt]);
      }
    }
#pragma unroll
    for (int nt = 0; nt < 4; ++nt)
#pragma unroll
      for (int v = 0; v < 8; ++v) {
        const int idx = (v + 8 * kh) * 64 + nt * 16 + row;
        TB[idx] = c[nt][v] * D2[idx];
      }
  }
  // g_t0 = (g_t2v @ m2^T) * silu'(t0)  -> D2
  {
    v8f c[4];
#pragma unroll
    for (int nt = 0; nt < 4; ++nt) c[nt] = zero8();
#pragma unroll
    for (int kc = 0; kc < 16; ++kc) {
      const int k0 = kc * 4 + kh * 2;
      v2f a; a.x = TB[row * 64 + k0]; a.y = TB[row * 64 + k0 + 1];
#pragma unroll
      for (int nt = 0; nt < 4; ++nt) {
        v2f b; b.x = m2[(nt * 16 + row) * 64 + k0]; b.y = m2[(nt * 16 + row) * 64 + k0 + 1];
        c[nt] = wmma4(a, b, c[nt]);
      }
    }
#pragma unroll
    for (int nt = 0; nt < 4; ++nt)
#pragma unroll
      for (int v = 0; v < 8; ++v) {
        const int idx = (v + 8 * kh) * 64 + nt * 16 + row;
        D2[idx] = c[nt][v] * D1[idx];
      }
  }
  // g_ef += g_t0 @ m1^T (64x8); cols 8..15 zero-padded
  {
    v8f c = zero8();
#pragma unroll
    for (int kc = 0; kc < 16; ++kc) {
      const int k0 = kc * 4 + kh * 2;
      v2f a; a.x = D2[row * 64 + k0]; a.y = D2[row * 64 + k0 + 1];
      v2f b;
      b.x = (row < 8) ? m1[row * 64 + k0] : 0.f;
      b.y = (row < 8) ? m1[row * 64 + k0 + 1] : 0.f;
      c = wmma4(a, b, c);
    }
#pragma unroll
    for (int v = 0; v < 8; ++v) {
      const int mr = v + 8 * kh;
      const long ge = e0 + mr;
      if (row < 8 && ge < E) gef[ge * 8 + row] += c[v];
    }
  }
}

// ---------------------------------------------------------------------------
// Geometry backward: (gY, gef) -> g_vec -> atomic force accumulation.
// ---------------------------------------------------------------------------
__global__ void k_geom_bwd(const float* __restrict__ vec,
                           const float* __restrict__ rr,
                           const float* __restrict__ gY,
                           const float* __restrict__ gef,
                           const int* __restrict__ snd,
                           const int* __restrict__ rcv,
                           float* __restrict__ forces, int E) {
  long e = (long)blockIdx.x * blockDim.x + threadIdx.x;
  if (e >= E) return;
  const float vx = vec[e * 3 + 0], vy = vec[e * 3 + 1], vz = vec[e * 3 + 2];
  const float r = rr[e];
  const float ir = 1.f / r;
  const float ux = vx * ir, uy = vy * ir, uz = vz * ir;
  const float s3 = 1.7320508075688772f, s5 = 2.2360679774997896f,
              s15 = 3.872983346207417f;
  float g[9];
#pragma unroll
  for (int m = 0; m < 9; ++m) g[m] = gY[e * 9 + m];
  // dY/du contracted with gY
  const float gux = s3 * g[1] + s15 * (uy * g[4] + uz * g[7] + ux * g[8]);
  const float guy = s3 * g[2] + s15 * (ux * g[4] + uz * g[5] - uy * g[8]);
  const float guz = s3 * g[3] + s15 * (uy * g[5] + ux * g[7]) + 3.f * s5 * uz * g[6];
  const float udg = ux * gux + uy * guy + uz * guz;
  float gvx = (gux - ux * udg) * ir;
  float gvy = (guy - uy * udg) * ir;
  float gvz = (guz - uz * udg) * ir;
  // radial part
  const float rm = fmaxf(r, 1e-6f);
  const float irm = 1.f / rm;
  const float drm = (r > 1e-6f) ? 1.f : 0.f;
  const float xc = r * 0.2f;
  float cut = 0.f, dcutdr = 0.f;
  if (xc < 1.f) {
    const float x2 = xc * xc, x4 = x2 * x2, x5 = x4 * xc, x6 = x5 * xc,
                x7 = x6 * xc;
    cut = 1.f - 21.f * x5 + 35.f * x6 - 15.f * x7;
    dcutdr = (-105.f * x4 + 210.f * x5 - 105.f * x6) * 0.2f;
  }
  const float pref = 0.6324555320336759f;
  const float PIf = 3.14159265358979323846f;
  float grs = 0.f;
#pragma unroll
  for (int nn = 1; nn <= 8; ++nn) {
    const float th = (float)nn * PIf * r * 0.2f;
    const float sn = sinf(th), cs = cosf(th);
    const float dth = (float)nn * PIf * 0.2f;
    const float b = pref * sn * irm;
    const float dbdr = pref * (dth * cs * irm - sn * irm * irm * drm);
    grs += gef[e * 8 + nn - 1] * (dbdr * cut + b * dcutdr);
  }
  gvx += grs * ux;
  gvy += grs * uy;
  gvz += grs * uz;
  const int s = snd[e], d = rcv[e];
  // forces = -dS/dpos; dS/dpos[recv]+=gvec, dS/dpos[send]-=gvec
  atomicAdd(&forces[(long)d * 3 + 0], -gvx);
  atomicAdd(&forces[(long)d * 3 + 1], -gvy);
  atomicAdd(&forces[(long)d * 3 + 2], -gvz);
  atomicAdd(&forces[(long)s * 3 + 0], gvx);
  atomicAdd(&forces[(long)s * 3 + 1], gvy);
  atomicAdd(&forces[(long)s * 3 + 2], gvz);
}

// ---------------------------------------------------------------------------
// Host-side orchestration
// ---------------------------------------------------------------------------
extern "C" void kernel_launch(void* const* d_in, const int* in_sizes, int n_in,
                              void* d_out, int out_size, void* d_ws,
                              size_t ws_size, hipStream_t stream) {
  const float* pos = (const float*)d_in[0];
  const float* attrs = (const float*)d_in[1];
  const float* shifts = (const float*)d_in[2];
  const float* Wemb = (const float*)d_in[3];
  const float* AE = (const float*)d_in[4];
  // per-layer params, sorted-key order:
  // 0 W_msg 1 W_prod 2 W_sc 3 W_up 4 m1 5 m2 6 m3 7 p1 8 p2 9 p3 10 r0 11 rA 12 rB
  const float* L[2][13];
  for (int l = 0; l < 2; ++l)
    for (int i = 0; i < 13; ++i) L[l][i] = (const float*)d_in[5 + l * 13 + i];
  const int* snd = (const int*)d_in[31];
  const int* rcv = (const int*)d_in[32];
  const int* batch = (const int*)d_in[33];
  const int E = in_sizes[31];
  const int N = in_sizes[33];
  const int Zn = in_sizes[1] / N;
  (void)n_in;

  float* w = (float*)d_ws;
  size_t off = 0;
  auto A_ = [&](size_t nelem) { float* p = w + off; off += nelem; return p; };
  float* vec = A_(3L * E);
  float* rr = A_(E);
  float* Ysh = A_(9L * E);
  float* efb = A_(8L * E);
  float* gY = A_(9L * E);
  float* gef = A_(8L * E);
  int* elem = (int*)A_(N);
  float* h0 = A_((long)N * KC);
  float* h1 = A_((long)N * KC);
  float* h2 = A_((long)N * KC);
  float* hup0 = A_((long)N * KC);
  float* hup1 = A_((long)N * KC);
  float* A0p = A_(9L * N * KC);
  float* A1p = A_(9L * N * KC);
  float* agg = A_(9L * N * KC);  // forward agg, reused as gagg in backward
  float* gA = A_(9L * N * KC);
  float* ghup = A_((long)N * KC);
  float* g_h = A_((long)N * KC);
  float* gout = A_((long)N * KC);
  if (off * sizeof(float) > ws_size) return;  // workspace too small

  float* out = (float*)d_out;
  float* totalO = out;           // [8]
  float* nodeE = out + 8;        // [N]
  float* forces = out + 8 + N;   // [N,3]
  hipMemsetAsync(d_out, 0, (size_t)out_size * sizeof(float), stream);

  const int nodeTiles = (N + 15) / 16;
  const int gemmBlocks = (nodeTiles + 7) / 8;   // 256 thr = 8 waves
  const int waveBlocks = (N + 7) / 8;           // wave-per-node, 256 thr
  const int edgeTiles = (E + 15) / 16;
  const int edgeBlocks = (edgeTiles + 1) / 2;   // 64 thr = 2 waves

  k_geom_fwd<<<(E + 255) / 256, 256, 0, stream>>>(pos, shifts, snd, rcv, vec,
                                                  rr, Ysh, efb, E);
  k_embed<<<(N + 255) / 256, 256, 0, stream>>>(attrs, Wemb, AE, elem, h0,
                                               nodeE, N, Zn);

  const float* hin[2] = {h0, h1};
  float* hout[2] = {h1, h2};
  float* hupL[2] = {hup0, hup1};
  float* ApL[2] = {A0p, A1p};
  for (int l = 0; l < 2; ++l) {
    hipMemsetAsync(agg, 0, 9L * N * KC * sizeof(float), stream);
    k_gemm32<<<gemmBlocks, 256, 0, stream>>>(hin[l], L[l][3], hupL[l], N, 0, 0);
    k_edge_fwd<<<edgeBlocks, 64, 0, stream>>>(efb, Ysh, hupL[l], snd, rcv,
                                              L[l][4], L[l][5], L[l][6], agg, E,
                                              N);
    k_gemm32_planes<<<gemmBlocks, 256, 0, stream>>>(agg, L[l][0], ApL[l], N, 0);
    if (l == 0)
      k_node_fwd<0><<<waveBlocks, 256, 0, stream>>>(
          hin[l], ApL[l], elem, L[l][2], L[l][1], L[l][7], L[l][8], L[l][9],
          L[l][10], L[l][11], L[l][12], hout[l], nodeE, N);
    else
      k_node_fwd<1><<<waveBlocks, 256, 0, stream>>>(
          hin[l], ApL[l], elem, L[l][2], L[l][1], L[l][7], L[l][8], L[l][9],
          L[l][10], L[l][11], L[l][12], hout[l], nodeE, N);
  }
  k_total<<<(N + 255) / 256, 256, 0, stream>>>(nodeE, batch, totalO, N);

  // ------------------------- backward -------------------------------------
  hipMemsetAsync(gY, 0, 9L * E * sizeof(float), stream);
  hipMemsetAsync(gef, 0, 8L * E * sizeof(float), stream);

  // layer 1
  k_readout1_bwd<<<waveBlocks, 256, 0, stream>>>(h2, L[1][11], L[1][12], gout, N);
  k_node_bwd<<<waveBlocks, 256, 0, stream>>>(gout, (const float*)nullptr, A1p,
                                             elem, L[1][2], L[1][1], L[1][7],
                                             L[1][8], L[1][9], gA, g_h, N, 1);
  k_gemm32_planes<<<gemmBlocks, 256, 0, stream>>>(gA, L[1][0], agg, N, 1);
  hipMemsetAsync(ghup, 0, (long)N * KC * sizeof(float), stream);
  k_edge_bwd<<<edgeBlocks, 64, 0, stream>>>(efb, Ysh, hup1, snd, rcv, L[1][4],
                                            L[1][5], L[1][6], agg, ghup, gY,
                                            gef, E, N, 1);
  k_gemm32<<<gemmBlocks, 256, 0, stream>>>(ghup, L[1][3], g_h, N, 1, 1);

  // layer 0 (gout0 = g_h + r0; input-h grad is discarded -> no ghup needed)
  k_node_bwd<<<waveBlocks, 256, 0, stream>>>(g_h, L[0][10], A0p, elem, L[0][2],
                                             L[0][1], L[0][7], L[0][8], L[0][9],
                                             gA, ghup, N, 0);
  k_gemm32_planes<<<gemmBlocks, 256, 0, stream>>>(gA, L[0][0], agg, N, 1);
  k_edge_bwd<<<edgeBlocks, 64, 0, stream>>>(efb, Ysh, hup0, snd, rcv, L[0][4],
                                            L[0][5], L[0][6], agg, ghup, gY,
                                            gef, E, N, 0);

  k_geom_bwd<<<(E + 255) / 256, 256, 0, stream>>>(vec, rr, gY, gef, snd, rcv,
                                                  forces, E);
}